// LightGCN_28286654611587
// MI455X (gfx1250) — compile-verified
//
#include <hip/hip_runtime.h>

#define NUM_USERS 100000
#define NUM_ITEMS 50000
#define N_NODES   150000   // NUM_USERS + NUM_ITEMS
#define EMBED_DIM 64
#define N_EDGES   2400000
#define NTOT      (N_NODES * EMBED_DIM)   // 9,600,000 floats
#define KEEP_SCALE 1.25f                  // 1/(1-0.2)
#define KEEP_THRESH 3435973836u           // 0.8 * 2^32

typedef __attribute__((ext_vector_type(2))) float v2f;
typedef __attribute__((ext_vector_type(8))) float v8f;

__device__ __forceinline__ unsigned lgcn_hash32(unsigned x) {
  x ^= x >> 16; x *= 0x7feb352dU;
  x ^= x >> 15; x *= 0x846ca68bU;
  x ^= x >> 16;
  return x;
}

// Fused concat + dropout; writes layer-0 embeddings and seeds the accumulator.
__global__ void lgcn_init_dropout(const float* __restrict__ ue,
                                  const float* __restrict__ ie,
                                  float* __restrict__ x0,
                                  float* __restrict__ acc) {
  int i = blockIdx.x * blockDim.x + threadIdx.x;
  if (i >= NTOT) return;
  float v = (i < NUM_USERS * EMBED_DIM) ? ue[i] : ie[i - NUM_USERS * EMBED_DIM];
  unsigned h = lgcn_hash32((unsigned)i ^ 0x9e3779b9U);
  float xv = (h < KEEP_THRESH) ? v * KEEP_SCALE : 0.0f;
  x0[i]  = xv;
  acc[i] = xv;
}

__global__ void lgcn_zero(float4* __restrict__ p, int n4) {
  int i = blockIdx.x * blockDim.x + threadIdx.x;
  if (i < n4) { float4 z; z.x = z.y = z.z = z.w = 0.0f; p[i] = z; }
}

// One wave32 per edge iteration: lane handles 2 of 64 dims.
// Coalesced 256B gather of x[col], 2x global_atomic_add_f32 scatter into y[row].
__global__ void lgcn_spmm(const int*   __restrict__ rows,
                          const int*   __restrict__ cols,
                          const float* __restrict__ vals,
                          const float* __restrict__ x,
                          float*       __restrict__ y,
                          int nwaves_total) {
  int gid  = blockIdx.x * blockDim.x + threadIdx.x;
  int wid  = gid >> 5;
  int lane = threadIdx.x & 31;
  for (int e = wid; e < N_EDGES; e += nwaves_total) {
    int   col = cols[e];
    int   row = rows[e];
    float val = vals[e];
    int en = e + nwaves_total;
    if (en < N_EDGES) {
      int coln = cols[en];
      __builtin_prefetch(&x[(size_t)coln * EMBED_DIM], 0, 3);  // global_prefetch_b8
    }
    const float2* src = reinterpret_cast<const float2*>(x + (size_t)col * EMBED_DIM);
    float2 g = src[lane];
    float* dst = y + (size_t)row * EMBED_DIM + lane * 2;
    unsafeAtomicAdd(dst,     g.x * val);   // global_atomic_add_f32 (L2-resident y)
    unsafeAtomicAdd(dst + 1, g.y * val);
  }
}

__global__ void lgcn_accum(float4* __restrict__ acc, const float4* __restrict__ y, int n4) {
  int i = blockIdx.x * blockDim.x + threadIdx.x;
  if (i >= n4) return;
  float4 a = acc[i];
  float4 b = y[i];
  a.x += b.x; a.y += b.y; a.z += b.z; a.w += b.w;
  acc[i] = a;
}

// out = 0.25*(acc + y), computed as D = (0.25*I)*AccTile + (0.25*I)*YTile
// with chained V_WMMA_F32_16X16X4_F32 over 16x16 tiles. One wave per tile.
// A (16x4 slice of 0.25*I): lanes0-15 hold K={0,1}, lanes16-31 K={2,3} (from immediates).
// B (4x16 slice): vgpr0 lanes0-15 -> row 4kk+0 (contiguous 64B), lanes16-31 -> row 4kk+2.
// C/D (16x16): 8 VGPRs, lanes0-15 M=vgpr, lanes16-31 M=vgpr+8.
__global__ void lgcn_final_wmma(const float* __restrict__ acc,
                                const float* __restrict__ y,
                                float* __restrict__ out) {
  const int total_tiles = (N_NODES / 16) * (EMBED_DIM / 16);  // 9375 * 4
  int gid = blockIdx.x * blockDim.x + threadIdx.x;
  int wid = gid >> 5;
  if (wid >= total_tiles) return;               // wave-uniform: EXEC stays all-1s
  int lane = threadIdx.x & 31;
  int rt = wid >> 2;
  int ct = wid & 3;
  int row0 = rt << 4;
  int col0 = ct << 4;
  int m  = lane & 15;
  int hi = lane >> 4;

  v8f d = {};
#pragma unroll
  for (int kk = 0; kk < 4; ++kk) {
    int kl = 4 * kk + 2 * hi;                   // local K row pair for this half-wave
    v2f a; a[0] = (kl     == m) ? 0.25f : 0.0f;
           a[1] = (kl + 1 == m) ? 0.25f : 0.0f;
    v2f b; b[0] = acc[(size_t)(row0 + kl    ) * EMBED_DIM + col0 + m];
           b[1] = acc[(size_t)(row0 + kl + 1) * EMBED_DIM + col0 + m];
    d = __builtin_amdgcn_wmma_f32_16x16x4_f32(false, a, false, b, (short)0, d, false, false);
  }
#pragma unroll
  for (int kk = 0; kk < 4; ++kk) {
    int kl = 4 * kk + 2 * hi;
    v2f a; a[0] = (kl     == m) ? 0.25f : 0.0f;
           a[1] = (kl + 1 == m) ? 0.25f : 0.0f;
    v2f b; b[0] = y[(size_t)(row0 + kl    ) * EMBED_DIM + col0 + m];
           b[1] = y[(size_t)(row0 + kl + 1) * EMBED_DIM + col0 + m];
    d = __builtin_amdgcn_wmma_f32_16x16x4_f32(false, a, false, b, (short)0, d, false, false);
  }
#pragma unroll
  for (int v = 0; v < 8; ++v)
    out[(size_t)(row0 + hi * 8 + v) * EMBED_DIM + col0 + m] = d[v];
}

extern "C" void kernel_launch(void* const* d_in, const int* in_sizes, int n_in,
                              void* d_out, int out_size, void* d_ws, size_t ws_size,
                              hipStream_t stream) {
  (void)in_sizes; (void)n_in; (void)out_size; (void)ws_size;
  const float* user_emb = (const float*)d_in[0];
  const float* item_emb = (const float*)d_in[1];
  const int*   adj_rows = (const int*)d_in[2];
  const int*   adj_cols = (const int*)d_in[3];
  const float* adj_vals = (const float*)d_in[4];
  float* out  = (float*)d_out;
  float* bufA = (float*)d_ws;                    // N*D floats
  float* bufB = bufA + (size_t)NTOT;             // N*D floats (ws >= 76.8 MB)

  const int B = 256;
  const int n4 = NTOT / 4;

  // layer-0 embeddings (with dropout) -> bufA, acc seeded in d_out
  lgcn_init_dropout<<<(NTOT + B - 1) / B, B, 0, stream>>>(user_emb, item_emb, bufA, out);

  const int SPMM_BLOCKS = 8192;                  // 65536 waves, ~37 edges each
  const int nwaves = SPMM_BLOCKS * (B / 32);

  const float* xs[3] = { bufA, bufB, bufA };
  float*       ys[3] = { bufB, bufA, bufB };

  for (int layer = 0; layer < 3; ++layer) {
    lgcn_zero<<<(n4 + B - 1) / B, B, 0, stream>>>((float4*)ys[layer], n4);
    lgcn_spmm<<<SPMM_BLOCKS, B, 0, stream>>>(adj_rows, adj_cols, adj_vals,
                                             xs[layer], ys[layer], nwaves);
    if (layer < 2)
      lgcn_accum<<<(n4 + B - 1) / B, B, 0, stream>>>((float4*)out, (const float4*)ys[layer], n4);
  }

  // out = 0.25 * (acc + y_layer3)  via WMMA (37500 tiles, 1 wave each)
  const int total_tiles = (N_NODES / 16) * (EMBED_DIM / 16);
  const int fin_threads = total_tiles * 32;
  lgcn_final_wmma<<<(fin_threads + B - 1) / B, B, 0, stream>>>(out, ys[2], out);
}